// transformerReg_3934190044189
// MI455X (gfx1250) — compile-verified
//
#include <hip/hip_runtime.h>
#include <hip/hip_bf16.h>

#define BN 2
#define NK 2048
#define CC 61
#define CF 64
#define DD 512
#define NL 2
#define LN_EPS 1e-5f

typedef __attribute__((ext_vector_type(2))) float v2f;
typedef __attribute__((ext_vector_type(8))) float v8f;

__device__ __forceinline__ v8f wmma4(v2f a, v2f b, v8f c) {
    // D = A(16x4,f32) * B(4x16,f32) + C(16x16,f32)
    return __builtin_amdgcn_wmma_f32_16x16x4_f32(
        /*neg_a=*/false, a, /*neg_b=*/false, b,
        /*c_mod=*/(short)0, c, /*reuse_a=*/false, /*reuse_b=*/false);
}

// ---------------------------------------------------------------------------
// f0 = concat(kpts, feats) along channel dim  -> (B, N, 64)
// ---------------------------------------------------------------------------
__global__ void k_concat(const float* __restrict__ kpts,
                         const float* __restrict__ feats,
                         float* __restrict__ f) {
    int idx = blockIdx.x * blockDim.x + threadIdx.x;
    if (idx >= BN * NK * CF) return;
    int c  = idx % CF;
    int bn = idx / CF;
    float v = (c < 3) ? kpts[bn * 3 + c] : feats[bn * CC + (c - 3)];
    f[idx] = v;
}

// ---------------------------------------------------------------------------
// broadcast disps (N, 512) to both batches of the d buffer
// ---------------------------------------------------------------------------
__global__ void k_dinit(const float* __restrict__ disps, float* __restrict__ d) {
    int idx = blockIdx.x * blockDim.x + threadIdx.x;
    if (idx >= NK * DD) return;
    float v = disps[idx];
    d[idx] = v;
    d[NK * DD + idx] = v;
}

// ---------------------------------------------------------------------------
// out[m][n] = relu( sum_k A[m][k] * W[n][k] + bias[n] )
// A: M x K row-major, W: N x K row-major, out: M x N row-major.
// One wave per 32x32 macro-tile (2x2 WMMA tiles): every A/B fragment feeds
// two WMMAs -> 8 FLOP/byte of global fragment traffic.
// Both fragment types are contiguous 8-byte loads per lane.
// ---------------------------------------------------------------------------
__global__ void k_linear_relu(const float* __restrict__ A,
                              const float* __restrict__ W,
                              const float* __restrict__ bias,
                              float* __restrict__ out,
                              int M, int N, int K) {
    int wave = (int)((blockIdx.x * blockDim.x + threadIdx.x) >> 5);
    int lane = threadIdx.x & 31;
    int ntn  = N >> 5;                          // 32-col macro tiles
    int tm   = (wave / ntn) << 5;
    int tn   = (wave % ntn) << 5;
    if (tm >= M) return;                        // wave-uniform

    int half = lane >> 4;                       // 0 | 1
    int l16  = lane & 15;
    int k0   = half << 1;                       // lane K offset within step-4

    const float* a0 = A + (size_t)(tm + l16) * K + k0;
    const float* a1 = a0 + (size_t)16 * K;
    const float* w0 = W + (size_t)(tn + l16) * K + k0;
    const float* w1 = w0 + (size_t)16 * K;

    v8f acc00 = {}, acc01 = {}, acc10 = {}, acc11 = {};
    for (int kk = 0; kk < K; kk += 4) {
        v2f A0 = *(const v2f*)(a0 + kk);
        v2f A1 = *(const v2f*)(a1 + kk);
        v2f B0 = *(const v2f*)(w0 + kk);
        v2f B1 = *(const v2f*)(w1 + kk);
        acc00 = wmma4(A0, B0, acc00);
        acc01 = wmma4(A0, B1, acc01);
        acc10 = wmma4(A1, B0, acc10);
        acc11 = wmma4(A1, B1, acc11);
    }

    float b0 = bias[tn + l16];
    float b1 = bias[tn + 16 + l16];
#pragma unroll
    for (int r = 0; r < 8; r++) {
        int m0 = tm + r + 8 * half;
        int m1 = m0 + 16;
        out[(size_t)m0 * N + tn + l16]      = fmaxf(acc00[r] + b0, 0.f);
        out[(size_t)m0 * N + tn + 16 + l16] = fmaxf(acc01[r] + b1, 0.f);
        out[(size_t)m1 * N + tn + l16]      = fmaxf(acc10[r] + b0, 0.f);
        out[(size_t)m1 * N + tn + 16 + l16] = fmaxf(acc11[r] + b1, 0.f);
    }
}

// ---------------------------------------------------------------------------
// LayerNorm over the full (N, CF) slab per batch: 3-stage deterministic tree.
// ---------------------------------------------------------------------------
__global__ void k_ln_partial(const float* __restrict__ x, float* __restrict__ part) {
    // grid = BN*64 blocks, 256 threads, 2048 floats per block
    __shared__ float ssum[256];
    __shared__ float ssq[256];
    int batch = blockIdx.x >> 6;
    int chunk = blockIdx.x & 63;
    const float4* p4 = (const float4*)(x + (size_t)batch * NK * CF + chunk * 2048);
    int tid = threadIdx.x;
    float s = 0.f, q = 0.f;
#pragma unroll
    for (int i = 0; i < 2; i++) {
        float4 v = p4[tid + i * 256];
        s += v.x + v.y + v.z + v.w;
        q += v.x * v.x + v.y * v.y + v.z * v.z + v.w * v.w;
    }
    ssum[tid] = s; ssq[tid] = q;
    __syncthreads();
    for (int off = 128; off > 0; off >>= 1) {
        if (tid < off) { ssum[tid] += ssum[tid + off]; ssq[tid] += ssq[tid + off]; }
        __syncthreads();
    }
    if (tid == 0) {
        part[blockIdx.x * 2 + 0] = ssum[0];
        part[blockIdx.x * 2 + 1] = ssq[0];
    }
}

__global__ void k_ln_finalize(const float* __restrict__ part, float* __restrict__ stats) {
    int b = threadIdx.x;
    if (b >= BN) return;
    float s = 0.f, q = 0.f;
    for (int i = 0; i < 64; i++) {
        s += part[(b * 64 + i) * 2 + 0];
        q += part[(b * 64 + i) * 2 + 1];
    }
    const float inv_n = 1.0f / (float)(NK * CF);
    float mu  = s * inv_n;
    float var = q * inv_n - mu * mu;
    stats[b * 2 + 0] = mu;
    stats[b * 2 + 1] = rsqrtf(var + LN_EPS);
}

__global__ void k_ln_apply(const float* __restrict__ x, const float* __restrict__ stats,
                           const float* __restrict__ gamma, const float* __restrict__ beta,
                           float* __restrict__ y) {
    int idx = blockIdx.x * blockDim.x + threadIdx.x;
    if (idx >= BN * NK * CF) return;
    int b  = idx / (NK * CF);
    int nc = idx % (NK * CF);
    y[idx] = (x[idx] - stats[b * 2]) * stats[b * 2 + 1] * gamma[nc] + beta[nc];
}

// ---------------------------------------------------------------------------
// sq[b][n] = sum_c f[b][n][c]^2
// ---------------------------------------------------------------------------
__global__ void k_rowsq(const float* __restrict__ f, float* __restrict__ sq) {
    int row = blockIdx.x * blockDim.x + threadIdx.x;
    if (row >= BN * NK) return;
    const float4* p = (const float4*)(f + (size_t)row * CF);
    float s = 0.f;
#pragma unroll
    for (int i = 0; i < CF / 4; i++) {
        float4 v = p[i];
        s += v.x * v.x + v.y * v.y + v.z * v.z + v.w * v.w;
    }
    sq[row] = s;
}

// ---------------------------------------------------------------------------
// Fused RBF attention:  out_i = sum_j exp(-||f_i - f_j||^2) d_j / rowsum_i
// Row max of -d2 is exactly 0 (d2_ii == 0), so no running-max is needed.
// Block = 8 waves; one 16-row tile per block; wave 0 builds the 16x16 score
// tile with WMMA and exponentiates it into LDS; all 8 waves then consume it
// as the WMMA A-matrix against their private 64-wide slice of d (D=512).
// ---------------------------------------------------------------------------
__global__ void k_attn(const float* __restrict__ f,   // B x N x CF (post-LN)
                       const float* __restrict__ sq,  // B x N
                       const float* __restrict__ d,   // B x N x DD
                       float* __restrict__ out) {     // B x N x DD
    __shared__ float s_S[16 * 16];
    __shared__ float s_rowsum[16];

    const int b    = blockIdx.x / (NK / 16);
    const int it   = blockIdx.x % (NK / 16);
    const int tid  = threadIdx.x;
    const int wave = tid >> 5;
    const int lane = tid & 31;
    const int half = lane >> 4;
    const int l16  = lane & 15;
    const int k0   = half << 1;

    const float* fb  = f  + (size_t)b * NK * CF;
    const float* sqb = sq + (size_t)b * NK;
    const float* db  = d  + (size_t)b * NK * DD;
    float*       ob  = out + (size_t)b * NK * DD;

    // wave 0 preloads the f_i tile as 16 A-fragments (K = 64 in steps of 4)
    v2f afrag[16];
    float sqi[8];
    if (wave == 0) {
        const float* firow = fb + (size_t)(it * 16 + l16) * CF + k0;
#pragma unroll
        for (int s = 0; s < 16; s++) afrag[s] = *(const v2f*)(firow + s * 4);
#pragma unroll
        for (int r = 0; r < 8; r++) sqi[r] = sqb[it * 16 + r + 8 * half];
    }

    if (tid < 16) s_rowsum[tid] = 0.f;
    __syncthreads();

    v8f acc[4] = {};                           // 16 rows x 64 cols per wave

    for (int jt = 0; jt < NK / 16; jt++) {
        // pull next tile toward the WGP while this one computes
        if (jt + 1 < NK / 16) {
            __builtin_prefetch(db + (size_t)((jt + 1) * 16 + l16) * DD
                                  + wave * 64 + half * 32, 0, 1);
            if (wave == 0)
                __builtin_prefetch(fb + (size_t)((jt + 1) * 16 + l16) * CF
                                      + half * 32, 0, 1);
        }

        if (wave == 0) {
            v8f sacc = {};
            const float* fjrow = fb + (size_t)(jt * 16 + l16) * CF + k0;
#pragma unroll
            for (int s = 0; s < 16; s++) {
                v2f bfrag = *(const v2f*)(fjrow + s * 4);
                sacc = wmma4(afrag[s], bfrag, sacc);
            }
            // sacc[r] = f_i[m] . f_j[n],  m = r + 8*half, n = l16
            float sqn = sqb[jt * 16 + l16];
#pragma unroll
            for (int r = 0; r < 8; r++) {
                float e = __expf(2.0f * sacc[r] - sqi[r] - sqn);
                s_S[(r + 8 * half) * 16 + l16] = e;
            }
        }
        __syncthreads();

        // accumulate exact row sums (threads 0..15)
        if (tid < 16) {
            float s = 0.f;
#pragma unroll
            for (int n2 = 0; n2 < 16; n2++) s += s_S[tid * 16 + n2];
            s_rowsum[tid] += s;
        }

        // A-fragments of S: load once, reuse across all 4 output tiles
        v2f sa[4];
#pragma unroll
        for (int ks = 0; ks < 4; ks++)
            sa[ks] = *(const v2f*)(&s_S[l16 * 16 + ks * 4 + k0]);

        // acc += S(16x16) * d_j(16x64-slice), K = 16 in 4 WMMA steps
#pragma unroll
        for (int t = 0; t < 4; t++) {
            int coln = wave * 64 + t * 16 + l16;
#pragma unroll
            for (int ks = 0; ks < 4; ks++) {
                const float* dp = db + (size_t)(jt * 16 + ks * 4 + k0) * DD + coln;
                v2f bb;
                bb.x = dp[0];
                bb.y = dp[DD];
                acc[t] = wmma4(sa[ks], bb, acc[t]);
            }
        }
        __syncthreads();                       // before s_S is overwritten
    }

    // normalize and store (rowsum complete: loop ended with a barrier)
#pragma unroll
    for (int t = 0; t < 4; t++) {
        int coln = wave * 64 + t * 16 + l16;
#pragma unroll
        for (int r = 0; r < 8; r++) {
            int m = r + 8 * half;
            ob[(size_t)(it * 16 + m) * DD + coln] = acc[t][r] / s_rowsum[m];
        }
    }
}

// ---------------------------------------------------------------------------
extern "C" void kernel_launch(void* const* d_in, const int* in_sizes, int n_in,
                              void* d_out, int out_size, void* d_ws, size_t ws_size,
                              hipStream_t stream) {
    const float* kpts  = (const float*)d_in[0];
    const float* disps = (const float*)d_in[1];
    const float* feats = (const float*)d_in[2];
    const float* Wf    = (const float*)d_in[3];
    const float* bf    = (const float*)d_in[4];
    const float* gamma = (const float*)d_in[5];
    const float* beta  = (const float*)d_in[6];
    const float* Wd    = (const float*)d_in[7];
    const float* bd    = (const float*)d_in[8];
    float* out = (float*)d_out;

    // workspace carve (floats)
    float* f_buf = (float*)d_ws;               // B*N*CF      = 262144
    float* f_lin = f_buf + (size_t)BN * NK * CF;
    float* d_A   = f_lin + (size_t)BN * NK * CF;   // B*N*DD = 2097152
    float* d_B   = d_A + (size_t)BN * NK * DD;
    float* sqv   = d_B + (size_t)BN * NK * DD;     // B*N
    float* part  = sqv + (size_t)BN * NK;          // 256
    float* stats = part + 256;                     // 4

    k_concat<<<(BN * NK * CF) / 256, 256, 0, stream>>>(kpts, feats, f_buf);
    k_dinit<<<(NK * DD) / 256, 256, 0, stream>>>(disps, d_A);

    const float* dcur = d_A;
    for (int l = 0; l < NL; l++) {
        // f = relu(f @ Wf^T + bf)   (32x32 macro-tiles, 8 waves/block)
        k_linear_relu<<<(BN * NK / 32) * (CF / 32) / 8, 256, 0, stream>>>(
            f_buf, Wf + (size_t)l * CF * CF, bf + (size_t)l * CF, f_lin,
            BN * NK, CF, CF);
        // LayerNorm over (N, CF) per batch
        k_ln_partial<<<BN * 64, 256, 0, stream>>>(f_lin, part);
        k_ln_finalize<<<1, 32, 0, stream>>>(part, stats);
        k_ln_apply<<<(BN * NK * CF) / 256, 256, 0, stream>>>(
            f_lin, stats, gamma + (size_t)l * NK * CF, beta + (size_t)l * NK * CF, f_buf);
        // sq rows of normalized f
        k_rowsq<<<(BN * NK) / 256, 256, 0, stream>>>(f_buf, sqv);
        // d = relu(d @ Wd^T + bd)
        k_linear_relu<<<(BN * NK / 32) * (DD / 32) / 8, 256, 0, stream>>>(
            dcur, Wd + (size_t)l * DD * DD, bd + (size_t)l * DD, d_B,
            BN * NK, DD, DD);
        // d = softmax(-d2) @ d   (fused, flash-style)
        float* dnext = (l == NL - 1) ? out : d_A;
        k_attn<<<BN * (NK / 16), 256, 0, stream>>>(f_buf, sqv, d_B, dnext);
        dcur = d_A;
    }
}